// ChamferLossSplit_81423989997793
// MI455X (gfx1250) — compile-verified
//
#include <hip/hip_runtime.h>
#include <math.h>

// ChamferLossSplit on MI455X (gfx1250, wave32).
// dist^2 = ||x||^2 + ||y||^2 - 2 x.y ; cross terms via V_WMMA_F32_16X16X4_F32.
// Two WMMAs per 16x16 tile (normal + transposed via operand swap) so both
// row-min and col-min are in-register reductions: no cross-lane shuffles.
// Masks folded into +inf norm entries: branch-free epilogue.
// Mins taken on squared distances (sqrt is monotonic); sqrt applied once
// per row/col at the end.

typedef __attribute__((ext_vector_type(2))) float v2f;
typedef __attribute__((ext_vector_type(8))) float v8f;

#define BB 256
#define NN 256
#define MM 256
#define DD 4

__device__ __forceinline__ float inf_f() { return __uint_as_float(0x7f800000u); }
__device__ __forceinline__ float fast_sqrtf(float x) { return __builtin_amdgcn_sqrtf(x); }

__global__ void __launch_bounds__(256)
chamfer_batch_kernel(const float* __restrict__ target,
                     const float* __restrict__ reco,
                     const int* __restrict__ in_pid,
                     const int* __restrict__ out_pid,
                     float* __restrict__ per_nonzero,
                     float* __restrict__ per_zero)
{
    __shared__ __align__(16) float s_t[NN * DD];     // target points
    __shared__ __align__(16) float s_r[MM * DD];     // reco points
    __shared__ __align__(16) float s_tna[NN];        // mask_x ? ||x||^2 : +inf
    __shared__ __align__(16) float s_rna[MM];        // mask_y ? ||y||^2 : +inf
    __shared__ unsigned s_minxy[NN];                 // per target row: min d^2 (bits)
    __shared__ unsigned s_minyx[MM];                 // per reco row:   min d^2 (bits)
    __shared__ float s_part[8 * 6];

    const int b    = blockIdx.x;
    const int tid  = threadIdx.x;         // 0..255
    const int wave = tid >> 5;
    const int lane = tid & 31;
    const int half = lane >> 4;           // 0: K=0,1   1: K=2,3
    const int lr   = lane & 15;

    // ---- stage batch into LDS; per-thread scalars stay in registers ----
    const float4 tv = reinterpret_cast<const float4*>(target + (size_t)b * NN * DD)[tid];
    const float4 rv = reinterpret_cast<const float4*>(reco   + (size_t)b * MM * DD)[tid];
    reinterpret_cast<float4*>(s_t)[tid] = tv;
    reinterpret_cast<float4*>(s_r)[tid] = rv;
    const float tn = tv.x * tv.x + tv.y * tv.y + tv.z * tv.z + tv.w * tv.w;
    const float rn = rv.x * rv.x + rv.y * rv.y + rv.z * rv.z + rv.w * rv.w;
    const int mx = (in_pid [b * NN + tid] != 0) ? 1 : 0;
    const int my = (out_pid[b * MM + tid] != 0) ? 1 : 0;
    s_tna[tid] = mx ? tn : inf_f();
    s_rna[tid] = my ? rn : inf_f();
    s_minxy[tid] = 0x7f800000u;
    s_minyx[tid] = 0x7f800000u;
    const float fx  = mx ? 1.0f : 0.0f;
    const float fy  = my ? 1.0f : 0.0f;
    const float xnc = mx ? fast_sqrtf(tn) : 0.0f;   // fallback numerator term
    const float ync = my ? 0.0f : fast_sqrtf(rn);   // zero-pid reco norm term
    __syncthreads();

    // ---- tile loop: each wave owns tiles t = wave, wave+8, ... (32 tiles) ----
    for (int t = wave; t < 256; t += 8) {
        const int i0 = (t >> 4) * 16;     // target tile origin
        const int j0 = (t & 15) * 16;     // reco tile origin

        // 32-bit 16x4 A layout: lanes 0-15 -> K=0,1 ; lanes 16-31 -> K=2,3
        v2f av, bv;
        av.x = s_t[(i0 + lr) * DD + 2 * half + 0];
        av.y = s_t[(i0 + lr) * DD + 2 * half + 1];
        bv.x = s_r[(j0 + lr) * DD + 2 * half + 0];
        bv.y = s_r[(j0 + lr) * DD + 2 * half + 1];

        // c1 = target x reco  (D[m_target][n_reco])
        // c2 = reco x target  (transposed tile, same LDS loads)
        v8f c1 = {}, c2 = {};
        c1 = __builtin_amdgcn_wmma_f32_16x16x4_f32(false, av, false, bv,
                                                   (short)0, c1, false, false);
        c2 = __builtin_amdgcn_wmma_f32_16x16x4_f32(false, bv, false, av,
                                                   (short)0, c2, false, false);

        // masked norms for the 8 rows this lane's accumulators cover
        const float4* t4 = reinterpret_cast<const float4*>(s_tna);
        const float4* r4 = reinterpret_cast<const float4*>(s_rna);
        const int ti4 = (i0 + 8 * half) >> 2;
        const int rj4 = (j0 + 8 * half) >> 2;
        const float4 ta = t4[ti4], tb = t4[ti4 + 1];
        const float4 ra = r4[rj4], rb = r4[rj4 + 1];
        const float tn8[8] = {ta.x, ta.y, ta.z, ta.w, tb.x, tb.y, tb.z, tb.w};
        const float rn8[8] = {ra.x, ra.y, ra.z, ra.w, rb.x, rb.y, rb.z, rb.w};
        const float rcol = s_rna[j0 + lr];   // column norm for c1
        const float tcol = s_tna[i0 + lr];   // column norm for c2

        float m1 = inf_f();   // min over target rows -> per-reco min (min_yx)
        float m2 = inf_f();   // min over reco rows   -> per-target min (min_xy)
        #pragma unroll
        for (int r = 0; r < 8; ++r) {
            m1 = fminf(m1, fmaxf(fmaf(-2.0f, c1[r], tn8[r] + rcol), 0.0f));
            m2 = fminf(m2, fmaxf(fmaf(-2.0f, c2[r], rn8[r] + tcol), 0.0f));
        }
        // values are >= 0 or +inf: unsigned bit-compare == float compare
        atomicMin(&s_minyx[j0 + lr], __float_as_uint(m1));
        atomicMin(&s_minxy[i0 + lr], __float_as_uint(m2));
    }
    __syncthreads();

    // ---- per-batch reductions (6 sums across 256 threads) ----
    float vals[6];
    vals[0] = fx;
    vals[1] = fy;
    vals[2] = mx ? fast_sqrtf(__uint_as_float(s_minxy[tid])) : 0.0f;  // sum_xy
    vals[3] = my ? fast_sqrtf(__uint_as_float(s_minyx[tid])) : 0.0f;  // sum_yx
    vals[4] = xnc;
    vals[5] = ync;

    #pragma unroll
    for (int k = 0; k < 6; ++k) {
        float v = vals[k];
        #pragma unroll
        for (int off = 16; off >= 1; off >>= 1)
            v += __shfl_xor(v, off, 32);
        vals[k] = v;
    }
    if (lane == 0) {
        #pragma unroll
        for (int k = 0; k < 6; ++k) s_part[wave * 6 + k] = vals[k];
    }
    __syncthreads();

    if (tid == 0) {
        float sum[6] = {0.f, 0.f, 0.f, 0.f, 0.f, 0.f};
        for (int w = 0; w < 8; ++w)
            for (int k = 0; k < 6; ++k) sum[k] += s_part[w * 6 + k];
        const float nx = sum[0], ny = sum[1];
        const float n_in  = fmaxf(1.0f, nx);
        const float n_out = fmaxf(1.0f, ny);
        const float normal   = 0.5f * (sum[2] / n_out + sum[3] / n_in);
        const float fallback = sum[4] / n_in;
        const float pnz = (ny == 0.0f) ? fallback : ((nx == 0.0f) ? 0.0f : normal);
        const float nyz = fmaxf(1.0f, (float)MM - ny);
        per_nonzero[b] = pnz;
        per_zero[b]    = sum[5] / nyz;
    }
}

__global__ void __launch_bounds__(256)
chamfer_finalize_kernel(const float* __restrict__ per_nonzero,
                        const float* __restrict__ per_zero,
                        float* __restrict__ out)
{
    __shared__ float sa[8], sb[8];
    const int tid = threadIdx.x;
    const int wave = tid >> 5, lane = tid & 31;
    float a = per_nonzero[tid];
    float c = per_zero[tid];
    #pragma unroll
    for (int off = 16; off >= 1; off >>= 1) {
        a += __shfl_xor(a, off, 32);
        c += __shfl_xor(c, off, 32);
    }
    if (lane == 0) { sa[wave] = a; sb[wave] = c; }
    __syncthreads();
    if (tid == 0) {
        float s1 = 0.f, s2 = 0.f;
        for (int w = 0; w < 8; ++w) { s1 += sa[w]; s2 += sb[w]; }
        out[0] = s1 / (float)BB;
        out[1] = s2 / (float)BB;
    }
}

extern "C" void kernel_launch(void* const* d_in, const int* in_sizes, int n_in,
                              void* d_out, int out_size, void* d_ws, size_t ws_size,
                              hipStream_t stream) {
    (void)in_sizes; (void)n_in; (void)out_size; (void)ws_size;
    const float* target  = (const float*)d_in[0];
    const float* reco    = (const float*)d_in[1];
    const int*   in_pid  = (const int*)d_in[2];
    const int*   out_pid = (const int*)d_in[3];
    float* out = (float*)d_out;

    float* ws_pnz = (float*)d_ws;        // [B]
    float* ws_pz  = ws_pnz + BB;         // [B]

    chamfer_batch_kernel<<<BB, 256, 0, stream>>>(target, reco, in_pid, out_pid,
                                                 ws_pnz, ws_pz);
    chamfer_finalize_kernel<<<1, 256, 0, stream>>>(ws_pnz, ws_pz, out);
}